// HeadC_24137716204089
// MI455X (gfx1250) — compile-verified
//
#include <hip/hip_runtime.h>
#include <math.h>

// Problem constants (from reference)
#define B_DIM   4096
#define T_DIM   50
#define C_DIM   600
#define H_DIM   100
#define CHUNK   6

// Padded tile dims for 16x16 WMMA tiling
#define TP 64    // T padded to 4 tiles of 16
#define HP 112   // H padded to 7 tiles of 16 (also LDS row stride for q/k/v)
#define SP 68    // row stride for the 64x64 score/probability board

typedef float v2f __attribute__((ext_vector_type(2)));
typedef float v8f __attribute__((ext_vector_type(8)));

__global__ __launch_bounds__(256)
void headc_fused_attn(const float* __restrict__ x,
                      const float* __restrict__ wq,
                      const float* __restrict__ wk,
                      const float* __restrict__ wv,
                      float* __restrict__ out) {
    __shared__ float sW[3 * H_DIM * CHUNK];   // staged weights
    __shared__ float sQ[TP * HP];
    __shared__ float sK[TP * HP];
    __shared__ float sV[TP * HP];
    __shared__ float sS[TP * SP];             // scores, then probabilities

    const int b      = blockIdx.x;
    const int tid    = threadIdx.x;
    const int lane   = tid & 31;              // wave32
    const int wave   = tid >> 5;
    const int nWaves = blockDim.x >> 5;       // 8

    // ---- stage grouped-conv weights in LDS (reused by all 5000 dots) ----
    for (int i = tid; i < H_DIM * CHUNK; i += blockDim.x) {
        sW[i]                     = wq[i];
        sW[H_DIM * CHUNK + i]     = wk[i];
        sW[2 * H_DIM * CHUNK + i] = wv[i];
    }
    __syncthreads();

    // ---- grouped 1x1 projection: one pass over x, q/k/v share the load ----
    // x chunk address = base + (t*600 + h*6) floats = 24-byte multiple from a
    // 256B-aligned base -> 8B aligned, so load as 3x float2 (global_load_b64).
    const float* xb = x + (size_t)b * T_DIM * C_DIM;
    for (int idx = tid; idx < TP * HP; idx += blockDim.x) {
        const int t = idx / HP;
        const int h = idx % HP;
        float q = 0.f, k = 0.f, v = 0.f;
        if (t < T_DIM && h < H_DIM) {
            const float2* xp2 = (const float2*)(xb + t * C_DIM + h * CHUNK);
            const float2 x01 = xp2[0];
            const float2 x23 = xp2[1];
            const float2 x45 = xp2[2];
            const float xv[CHUNK] = {x01.x, x01.y, x23.x, x23.y, x45.x, x45.y};
            const float* wqp = sW + h * CHUNK;
            const float* wkp = sW + H_DIM * CHUNK + h * CHUNK;
            const float* wvp = sW + 2 * H_DIM * CHUNK + h * CHUNK;
            #pragma unroll
            for (int c = 0; c < CHUNK; ++c) {
                q = fmaf(xv[c], wqp[c], q);
                k = fmaf(xv[c], wkp[c], k);
                v = fmaf(xv[c], wvp[c], v);
            }
        }
        sQ[idx] = q;   // zero padding outside [50,100) makes WMMA tiling exact
        sK[idx] = k;
        sV[idx] = v;
    }
    __syncthreads();

    // Per-lane fragment geometry (ISA 7.12.2 f32 layouts, wave32):
    //   A 16x4 : lanes 0-15 hold K={0,1}, lanes 16-31 hold K={2,3}; row = M.
    //   B 4x16 : mirrored; col = N.
    //   C/D    : VGPR i -> rows M=i (lanes 0-15) and M=8+i (lanes 16-31).
    const int row16 = lane & 15;
    const int khalf = (lane >> 4) << 1;   // 0 or 2

    // ---- S = Q @ K^T : 4x4 tiles of 16x16, K = 100 in 25 steps of 4 ----
    for (int tIdx = wave; tIdx < 16; tIdx += nWaves) {
        const int m0 = (tIdx >> 2) << 4;
        const int n0 = (tIdx & 3) << 4;
        v8f acc = {};
        for (int k0 = 0; k0 < H_DIM; k0 += 4) {
            v2f a, bf;
            const float* ap = sQ + (m0 + row16) * HP + k0 + khalf;
            a.x = ap[0]; a.y = ap[1];
            // B = K^T  =>  B(kk, n) = K[n][kk]
            const float* bp = sK + (n0 + row16) * HP + k0 + khalf;
            bf.x = bp[0]; bf.y = bp[1];
            acc = __builtin_amdgcn_wmma_f32_16x16x4_f32(
                false, a, false, bf, (short)0, acc, false, false);
        }
        float* dst = sS + (m0 + ((lane >> 4) << 3)) * SP + n0 + row16;
        #pragma unroll
        for (int i = 0; i < 8; ++i) dst[i * SP] = acc[i];
    }
    __syncthreads();

    // ---- causal softmax, one thread per query row ----
    const float scale = rsqrtf((float)C_DIM);   // C^-0.5 per reference
    if (tid < TP) {
        const int t = tid;
        float* rowp = sS + t * SP;
        if (t < T_DIM) {
            float mx = -INFINITY;
            for (int j = 0; j <= t; ++j) mx = fmaxf(mx, rowp[j] * scale);
            float sum = 0.f;
            for (int j = 0; j <= t; ++j) {
                const float e = __expf(rowp[j] * scale - mx);
                rowp[j] = e;
                sum += e;
            }
            const float inv = 1.f / sum;
            for (int j = 0; j <= t; ++j) rowp[j] *= inv;
            for (int j = t + 1; j < TP; ++j) rowp[j] = 0.f;  // causal + pad
        } else {
            for (int j = 0; j < TP; ++j) rowp[j] = 0.f;      // pad rows
        }
    }
    __syncthreads();

    // ---- O = P @ V : 4x7 tiles, K = 64 in 16 steps; write straight out ----
    float* ob = out + (size_t)b * T_DIM * H_DIM;
    for (int tIdx = wave; tIdx < 4 * 7; tIdx += nWaves) {
        const int m0 = (tIdx / 7) << 4;
        const int n0 = (tIdx % 7) << 4;
        v8f acc = {};
        for (int k0 = 0; k0 < TP; k0 += 4) {
            v2f a, bf;
            const float* ap = sS + (m0 + row16) * SP + k0 + khalf;
            a.x = ap[0]; a.y = ap[1];
            // B = V  =>  B(kk, n) = V[kk][n]
            const float* bp = sV + (k0 + khalf) * HP + n0 + row16;
            bf.x = bp[0]; bf.y = bp[HP];
            acc = __builtin_amdgcn_wmma_f32_16x16x4_f32(
                false, a, false, bf, (short)0, acc, false, false);
        }
        const int h     = n0 + row16;
        const int mbase = m0 + ((lane >> 4) << 3);
        if (h < H_DIM) {
            #pragma unroll
            for (int i = 0; i < 8; ++i) {
                const int t = mbase + i;
                if (t < T_DIM) ob[t * H_DIM + h] = acc[i];
            }
        }
    }
}

extern "C" void kernel_launch(void* const* d_in, const int* in_sizes, int n_in,
                              void* d_out, int out_size, void* d_ws, size_t ws_size,
                              hipStream_t stream) {
    (void)in_sizes; (void)n_in; (void)out_size; (void)d_ws; (void)ws_size;
    const float* x  = (const float*)d_in[0];
    const float* wq = (const float*)d_in[1];
    const float* wk = (const float*)d_in[2];
    const float* wv = (const float*)d_in[3];
    float* out = (float*)d_out;
    headc_fused_attn<<<dim3(B_DIM), dim3(256), 0, stream>>>(x, wq, wk, wv, out);
}